// DeformableUpsampleBlock_75204877353407
// MI455X (gfx1250) — compile-verified
//
#include <hip/hip_runtime.h>

// ---------------------------------------------------------------------------
// DeformableUpsampleBlock on gfx1250 (MI455X).
// fp32 path throughout (matches reference dtype); GEMMs use
// V_WMMA_F32_16X16X4_F32. One wave32 per 16-pixel tile; 4 waves per block.
// Branchless clamped gathers (unconditional load + cndmask), incremental
// (c,t) tracking instead of per-iteration div/mod, b64 weight loads.
// ---------------------------------------------------------------------------

typedef __attribute__((ext_vector_type(2))) float v2f;
typedef __attribute__((ext_vector_type(8))) float v8f;

static constexpr int BATCH = 8;
static constexpr int CIN   = 64;
static constexpr int CMID  = 64;
static constexpr int CCAT  = 128;
static constexpr int COUT  = 32;
static constexpr int H     = 64;
static constexpr int W     = 64;
static constexpr int HWs   = H * W;          // 4096
static constexpr int NPIX  = BATCH * HWs;    // 32768
static constexpr int K3    = CIN * 9;        // 576
static constexpr float EPS = 1e-5f;

__device__ __forceinline__ v8f wmma_f32(v2f a, v2f b, v8f c) {
  // 8-arg form: (neg_a, A, neg_b, B, c_mod, C, reuse_a, reuse_b)
  return __builtin_amdgcn_wmma_f32_16x16x4_f32(false, a, false, b, (short)0, c,
                                               false, false);
}

__device__ __forceinline__ v2f load2(const float* __restrict__ p) {
  return *reinterpret_cast<const v2f*>(p);   // 8B-aligned by construction
}

// Branchless bilinear sample, zeros outside [0,H)x[0,W) (matches reference).
// Loads are unconditional from clamped (always valid) addresses; the OOB
// zeroing is a post-load select so no EXEC manipulation is generated.
__device__ __forceinline__ float bilin64(const float* __restrict__ plane,
                                         float py, float px) {
  float fy = floorf(py), fx = floorf(px);
  float wy = py - fy, wx = px - fx;
  int yi = (int)fy, xi = (int)fx;
  int y0 = min(max(yi, 0), H - 1), y1 = min(max(yi + 1, 0), H - 1);
  int x0 = min(max(xi, 0), W - 1), x1 = min(max(xi + 1, 0), W - 1);
  float v00 = plane[y0 * W + x0];
  float v01 = plane[y0 * W + x1];
  float v10 = plane[y1 * W + x0];
  float v11 = plane[y1 * W + x1];
  bool y0ok = ((unsigned)yi       < (unsigned)H);
  bool y1ok = ((unsigned)(yi + 1) < (unsigned)H);
  bool x0ok = ((unsigned)xi       < (unsigned)W);
  bool x1ok = ((unsigned)(xi + 1) < (unsigned)W);
  v00 = (y0ok && x0ok) ? v00 : 0.f;
  v01 = (y0ok && x1ok) ? v01 : 0.f;
  v10 = (y1ok && x0ok) ? v10 : 0.f;
  v11 = (y1ok && x1ok) ? v11 : 0.f;
  return v00 * (1.f - wy) * (1.f - wx) + v01 * (1.f - wy) * wx +
         v10 * wy * (1.f - wx) + v11 * wy * wx;
}

// ---- kernel 1: BN+ReLU over x -> h ----------------------------------------
__global__ void bn_relu3_kernel(const float* __restrict__ x,
                                const float* __restrict__ g,
                                const float* __restrict__ be,
                                const float* __restrict__ mu,
                                const float* __restrict__ var,
                                float* __restrict__ h) {
  int idx = blockIdx.x * blockDim.x + threadIdx.x;
  if (idx >= BATCH * CIN * HWs) return;
  int c = (idx >> 12) & (CIN - 1);
  float s = g[c] * rsqrtf(var[c] + EPS);
  h[idx] = fmaxf((x[idx] - mu[c]) * s + be[c], 0.f);
}

// ---- kernel 2: 3x3 offset conv (implicit GEMM, WMMA f32) -------------------
// M=16 pixels/tile, N=32 (channels 18..31 padded with zero weights), K=576.
__global__ __launch_bounds__(128) void conv_off3_kernel(
    const float* __restrict__ h, const float* __restrict__ w,
    const float* __restrict__ bias, float* __restrict__ off3) {
  __shared__ int s_ky[9];
  __shared__ int s_kx[9];
  if (threadIdx.x < 9) {
    s_ky[threadIdx.x] = (int)threadIdx.x / 3 - 1;
    s_kx[threadIdx.x] = (int)threadIdx.x % 3 - 1;
  }
  __syncthreads();
  const int wave = threadIdx.x >> 5;
  const int lane = threadIdx.x & 31;
  const int pix0 = (blockIdx.x * 4 + wave) * 16;
  const int b  = pix0 >> 12;
  const int y  = (pix0 >> 6) & (H - 1);
  const int x0 = pix0 & (W - 1);
  const int m  = lane & 15;
  const int hi = lane >> 4;
  const int n  = lane & 15;
  const int n1c = min(n + 16, 17);             // clamped row for padded tile
  const bool n1ok = (n + 16) < 18;
  const float* hb = h + (size_t)b * CIN * HWs;
  const float* w0p = w + n * K3 + 2 * hi;      // row n,  this lane's k base
  const float* w1p = w + n1c * K3 + 2 * hi;    // row n+16 (clamped)
  __builtin_prefetch(w0p, 0, 1);
  v8f acc0 = {}, acc1 = {};
  int ca = 0, ta = 2 * hi;                     // (channel, tap) of lane's ka
  for (int k0 = 0; k0 < K3; k0 += 4) {
    // A element 0: k = k0 + 2*hi  -> (ca, ta)
    int yy0 = y + s_ky[ta];
    int xx0 = x0 + m + s_kx[ta];
    bool ok0 = ((unsigned)yy0 < (unsigned)H) && ((unsigned)xx0 < (unsigned)W);
    float a0 = hb[ca * HWs + min(max(yy0, 0), H - 1) * W + min(max(xx0, 0), W - 1)];
    // A element 1: k+1 -> (cb, tb)
    int tb = ta + 1, cb = ca;
    if (tb == 9) { tb = 0; ++cb; }
    int yy1 = y + s_ky[tb];
    int xx1 = x0 + m + s_kx[tb];
    bool ok1 = ((unsigned)yy1 < (unsigned)H) && ((unsigned)xx1 < (unsigned)W);
    float a1 = hb[cb * HWs + min(max(yy1, 0), H - 1) * W + min(max(xx1, 0), W - 1)];
    v2f a; a.x = ok0 ? a0 : 0.f; a.y = ok1 ? a1 : 0.f;
    v2f b0 = load2(w0p + k0);
    v2f b1 = load2(w1p + k0);
    b1.x = n1ok ? b1.x : 0.f;
    b1.y = n1ok ? b1.y : 0.f;
    acc0 = wmma_f32(a, b0, acc0);
    acc1 = wmma_f32(a, b1, acc1);
    ta += 4; if (ta >= 9) { ta -= 9; ++ca; }
  }
#pragma unroll
  for (int j = 0; j < 8; ++j) {
    int xx = x0 + j + 8 * hi;
    off3[(((size_t)b * 18 + n) * H + y) * W + xx] = acc0[j] + bias[n];
    if (n1ok)
      off3[(((size_t)b * 18 + n + 16) * H + y) * W + xx] = acc1[j] + bias[n + 16];
  }
}

// ---- kernel 3: 3x3 deformable conv (sampled implicit GEMM, WMMA f32) -------
// M=16 pixels, N=64 (4 tiles), K=576; sampling coords staged in LDS.
__global__ __launch_bounds__(128) void deform3_kernel(
    const float* __restrict__ h, const float* __restrict__ off3,
    const float* __restrict__ w, const float* __restrict__ bias,
    float* __restrict__ out3) {
  __shared__ float s_py[4][16 * 9];
  __shared__ float s_px[4][16 * 9];
  const int wave = threadIdx.x >> 5;
  const int lane = threadIdx.x & 31;
  const int pix0 = (blockIdx.x * 4 + wave) * 16;
  const int b  = pix0 >> 12;
  const int y  = (pix0 >> 6) & (H - 1);
  const int x0 = pix0 & (W - 1);
  for (int i = lane; i < 16 * 9; i += 32) {
    int mm = i / 9, t = i % 9;
    int xx = x0 + mm;
    float dy = off3[(((size_t)b * 18 + 2 * t + 0) * H + y) * W + xx];
    float dx = off3[(((size_t)b * 18 + 2 * t + 1) * H + y) * W + xx];
    s_py[wave][i] = (float)(y + t / 3 - 1) + dy;
    s_px[wave][i] = (float)(xx + t % 3 - 1) + dx;
  }
  __syncthreads();
  const int m  = lane & 15;
  const int hi = lane >> 4;
  const int n  = lane & 15;
  const float* hb = h + (size_t)b * CIN * HWs;
  const float* wp = w + n * K3 + 2 * hi;       // row n, this lane's k base
  __builtin_prefetch(wp, 0, 1);
  const float* pyb = &s_py[wave][m * 9];
  const float* pxb = &s_px[wave][m * 9];
  v8f acc0 = {}, acc1 = {}, acc2 = {}, acc3 = {};
  int ca = 0, ta = 2 * hi;
  for (int k0 = 0; k0 < K3; k0 += 4) {
    v2f a;
    a.x = bilin64(hb + (size_t)ca * HWs, pyb[ta], pxb[ta]);
    int tb = ta + 1, cb = ca;
    if (tb == 9) { tb = 0; ++cb; }
    a.y = bilin64(hb + (size_t)cb * HWs, pyb[tb], pxb[tb]);
    v2f b0 = load2(wp +  0 * K3 + k0);
    v2f b1 = load2(wp + 16 * K3 + k0);
    v2f b2 = load2(wp + 32 * K3 + k0);
    v2f b3 = load2(wp + 48 * K3 + k0);
    acc0 = wmma_f32(a, b0, acc0);
    acc1 = wmma_f32(a, b1, acc1);
    acc2 = wmma_f32(a, b2, acc2);
    acc3 = wmma_f32(a, b3, acc3);
    ta += 4; if (ta >= 9) { ta -= 9; ++ca; }
  }
#pragma unroll
  for (int j = 0; j < 8; ++j) {
    int xx = x0 + j + 8 * hi;
    out3[(((size_t)b * CMID + n +  0) * H + y) * W + xx] = acc0[j] + bias[n +  0];
    out3[(((size_t)b * CMID + n + 16) * H + y) * W + xx] = acc1[j] + bias[n + 16];
    out3[(((size_t)b * CMID + n + 32) * H + y) * W + xx] = acc2[j] + bias[n + 32];
    out3[(((size_t)b * CMID + n + 48) * H + y) * W + xx] = acc3[j] + bias[n + 48];
  }
}

// ---- kernel 4: BN+ReLU over virtual concat [x, out3] -> h1 -----------------
__global__ void bn_relu_cat_kernel(const float* __restrict__ x,
                                   const float* __restrict__ out3,
                                   const float* __restrict__ g,
                                   const float* __restrict__ be,
                                   const float* __restrict__ mu,
                                   const float* __restrict__ var,
                                   float* __restrict__ h1) {
  int idx = blockIdx.x * blockDim.x + threadIdx.x;
  if (idx >= BATCH * CCAT * HWs) return;
  int p = idx & (HWs - 1);
  int c = (idx >> 12) & (CCAT - 1);
  int b = idx >> 19;
  float src = (c < CIN) ? x[((size_t)(b * CIN + c)) * HWs + p]
                        : out3[((size_t)(b * CMID + (c - CIN))) * HWs + p];
  float s = g[c] * rsqrtf(var[c] + EPS);
  h1[idx] = fmaxf((src - mu[c]) * s + be[c], 0.f);
}

// ---- kernel 5: 1x1 offset conv (2 outputs) ---------------------------------
__global__ void conv_off1_kernel(const float* __restrict__ h1,
                                 const float* __restrict__ w,
                                 const float* __restrict__ bias,
                                 float* __restrict__ off1) {
  int pix = blockIdx.x * blockDim.x + threadIdx.x;
  if (pix >= NPIX) return;
  int b = pix >> 12;
  int p = pix & (HWs - 1);
  const float* hp = h1 + (size_t)b * CCAT * HWs + p;
  float s0 = bias[0], s1 = bias[1];
  for (int c = 0; c < CCAT; ++c) {
    float v = hp[(size_t)c * HWs];
    s0 += w[c] * v;
    s1 += w[CCAT + c] * v;
  }
  off1[((size_t)b * 2 + 0) * HWs + p] = s0;
  off1[((size_t)b * 2 + 1) * HWs + p] = s1;
}

// ---- kernel 6: 1x1 deformable conv + nearest 2x upsample (WMMA f32) --------
// M=16 pixels, N=32 (2 tiles), K=128.
__global__ __launch_bounds__(128) void deform1_up_kernel(
    const float* __restrict__ h1, const float* __restrict__ off1,
    const float* __restrict__ w, const float* __restrict__ bias,
    float* __restrict__ out) {
  __shared__ float s_py[4][16];
  __shared__ float s_px[4][16];
  const int wave = threadIdx.x >> 5;
  const int lane = threadIdx.x & 31;
  const int pix0 = (blockIdx.x * 4 + wave) * 16;
  const int b  = pix0 >> 12;
  const int y  = (pix0 >> 6) & (H - 1);
  const int x0 = pix0 & (W - 1);
  if (lane < 16) {
    int xx = x0 + lane;
    float dy = off1[(((size_t)b * 2 + 0) * H + y) * W + xx];
    float dx = off1[(((size_t)b * 2 + 1) * H + y) * W + xx];
    s_py[wave][lane] = (float)y + dy;
    s_px[wave][lane] = (float)xx + dx;
  }
  __syncthreads();
  const int m  = lane & 15;
  const int hi = lane >> 4;
  const int n  = lane & 15;
  const float* hb = h1 + (size_t)b * CCAT * HWs;
  const float* wp = w + n * CCAT + 2 * hi;
  __builtin_prefetch(wp, 0, 1);
  float py = s_py[wave][m], px = s_px[wave][m];
  v8f acc0 = {}, acc1 = {};
  for (int k0 = 0; k0 < CCAT; k0 += 4) {
    const int ka = k0 + 2 * hi;
    v2f a;
    a.x = bilin64(hb + (size_t)(ka + 0) * HWs, py, px);
    a.y = bilin64(hb + (size_t)(ka + 1) * HWs, py, px);
    v2f b0 = load2(wp +  0 * CCAT + k0);
    v2f b1 = load2(wp + 16 * CCAT + k0);
    acc0 = wmma_f32(a, b0, acc0);
    acc1 = wmma_f32(a, b1, acc1);
  }
#pragma unroll
  for (int j = 0; j < 8; ++j) {
    int xx = x0 + j + 8 * hi;
    float v0 = acc0[j] + bias[n];
    float v1 = acc1[j] + bias[n + 16];
    size_t o0 = (((size_t)b * COUT + n +  0) * (2 * H) + 2 * y) * (2 * W) + 2 * xx;
    size_t o1 = (((size_t)b * COUT + n + 16) * (2 * H) + 2 * y) * (2 * W) + 2 * xx;
    out[o0] = v0; out[o0 + 1] = v0; out[o0 + 2 * W] = v0; out[o0 + 2 * W + 1] = v0;
    out[o1] = v1; out[o1 + 1] = v1; out[o1 + 2 * W] = v1; out[o1 + 2 * W + 1] = v1;
  }
}

// ---------------------------------------------------------------------------
extern "C" void kernel_launch(void* const* d_in, const int* in_sizes, int n_in,
                              void* d_out, int out_size, void* d_ws, size_t ws_size,
                              hipStream_t stream) {
  (void)in_sizes; (void)n_in; (void)out_size; (void)ws_size;
  const float* x         = (const float*)d_in[0];
  const float* bn3_gamma = (const float*)d_in[1];
  const float* bn3_beta  = (const float*)d_in[2];
  const float* bn3_mean  = (const float*)d_in[3];
  const float* bn3_var   = (const float*)d_in[4];
  const float* w_off3    = (const float*)d_in[5];
  const float* b_off3    = (const float*)d_in[6];
  const float* w_d3      = (const float*)d_in[7];
  const float* b_d3      = (const float*)d_in[8];
  const float* bn1_gamma = (const float*)d_in[9];
  const float* bn1_beta  = (const float*)d_in[10];
  const float* bn1_mean  = (const float*)d_in[11];
  const float* bn1_var   = (const float*)d_in[12];
  const float* w_off1    = (const float*)d_in[13];
  const float* b_off1    = (const float*)d_in[14];
  const float* w_d1      = (const float*)d_in[15];
  const float* b_d1      = (const float*)d_in[16];
  float* out = (float*)d_out;

  // workspace layout (floats): ~36.2 MB total
  float* ws    = (float*)d_ws;
  float* h     = ws;                 // 8*64*4096   = 2,097,152
  float* off3  = ws + 2097152;       // 8*18*4096   =   589,824
  float* out3  = ws + 2686976;       // 8*64*4096   = 2,097,152
  float* h1    = ws + 4784128;       // 8*128*4096  = 4,194,304
  float* off1  = ws + 8978432;       // 8*2*4096    =    65,536

  const int TILES  = NPIX / 16;      // 2048 tiles of 16 pixels
  const int BLOCKS = TILES / 4;      // 4 waves (tiles) per 128-thread block

  bn_relu3_kernel<<<(BATCH * CIN * HWs + 255) / 256, 256, 0, stream>>>(
      x, bn3_gamma, bn3_beta, bn3_mean, bn3_var, h);
  conv_off3_kernel<<<BLOCKS, 128, 0, stream>>>(h, w_off3, b_off3, off3);
  deform3_kernel<<<BLOCKS, 128, 0, stream>>>(h, off3, w_d3, b_d3, out3);
  bn_relu_cat_kernel<<<(BATCH * CCAT * HWs + 255) / 256, 256, 0, stream>>>(
      x, out3, bn1_gamma, bn1_beta, bn1_mean, bn1_var, h1);
  conv_off1_kernel<<<(NPIX + 255) / 256, 256, 0, stream>>>(h1, w_off1, b_off1, off1);
  deform1_up_kernel<<<BLOCKS, 128, 0, stream>>>(h1, off1, w_d1, b_d1, out);
}